// DynamicOTThresh_41790031790463
// MI455X (gfx1250) — compile-verified
//
#include <hip/hip_runtime.h>
#include <hip/hip_bf16.h>

#define BNUM 256
#define NELEM 65536
#define NV4 (NELEM / 4)      // 16384 float4 per row
#define BINS 256
#define NROWS_TOT 512        // 2 matrices * 256 rows

typedef float v2f __attribute__((ext_vector_type(2)));
typedef float v8f __attribute__((ext_vector_type(8)));

// ---------------------------------------------------------------------------
// Kernel 1: per-row min/max then 256-bin histogram (one block per (mat,row)).
// ---------------------------------------------------------------------------
__global__ void row_minmax_hist(const float* __restrict__ A,
                                const float* __restrict__ Bm,
                                float* __restrict__ mnmx,   // [512][2]
                                float* __restrict__ hist)   // [512][256]
{
    const int bid = blockIdx.x;          // 0..511
    const int mat = bid >> 8;            // 0 or 1
    const int row = bid & 255;
    const int tid = threadIdx.x;         // 0..255

    const float* xrow = ((mat == 0) ? A : Bm) + (size_t)row * NELEM;
    const float4* x4  = (const float4*)xrow;

    // ---- pass 1: min / max ----
    float mn =  __builtin_huge_valf();
    float mx = -__builtin_huge_valf();
    for (int i = tid; i < NV4; i += 256) {
        float4 v = x4[i];
        mn = fminf(mn, fminf(fminf(v.x, v.y), fminf(v.z, v.w)));
        mx = fmaxf(mx, fmaxf(fmaxf(v.x, v.y), fmaxf(v.z, v.w)));
    }
    __shared__ float smn[256];
    __shared__ float smx[256];
    smn[tid] = mn; smx[tid] = mx;
    __syncthreads();
    for (int s = 128; s > 0; s >>= 1) {
        if (tid < s) {
            smn[tid] = fminf(smn[tid], smn[tid + s]);
            smx[tid] = fmaxf(smx[tid], smx[tid + s]);
        }
        __syncthreads();
    }
    const float rowmn = smn[0];
    const float rowmx = smx[0];
    const float scale = (rowmx > rowmn) ? (rowmx - rowmn) : 1.0f;

    // ---- pass 2: histogram (LDS uint atomics) ----
    __shared__ unsigned int shist[BINS];
    shist[tid] = 0u;
    __syncthreads();
    for (int i = tid; i < NV4; i += 256) {
        float4 v = x4[i];
        #pragma unroll
        for (int c = 0; c < 4; ++c) {
            float xv = (c == 0) ? v.x : (c == 1) ? v.y : (c == 2) ? v.z : v.w;
            float t  = (xv - rowmn) / scale * 256.0f;   // same op order as reference
            int b = (int)floorf(t);
            b = (b < 0) ? 0 : ((b > 255) ? 255 : b);
            atomicAdd(&shist[b], 1u);
        }
    }
    __syncthreads();
    hist[(size_t)bid * BINS + tid] = (float)shist[tid];
    if (tid == 0) {
        mnmx[bid * 2 + 0] = rowmn;
        mnmx[bid * 2 + 1] = rowmx;
    }
}

// ---------------------------------------------------------------------------
// Kernel 2: dual cumulative sums via f32 WMMA GEMM against implicit upper-
// triangular ones matrix.  omega_raw = hist @ U,  mu_raw = (hist .* k) @ U.
// One wave (32 lanes) computes one 16x16 tile of both outputs.
// ---------------------------------------------------------------------------
__global__ void otsu_cumsum_wmma(const float* __restrict__ hist,  // [512][256]
                                 float* __restrict__ omega,       // [512][256]
                                 float* __restrict__ mu)          // [512][256]
{
    const int lane  = threadIdx.x;       // 0..31
    const int ntile = blockIdx.x;        // 0..15  (bin columns)
    const int mtile = blockIdx.y;        // 0..31  (row tiles)
    const int n0 = ntile * 16;
    const int m0 = mtile * 16;

    const int hi   = lane >> 4;          // 0: K=0,1 ; 1: K=2,3
    const int l15  = lane & 15;
    const int arow = m0 + l15;           // A-matrix row held by this lane
    const int bcol = n0 + l15;           // B/C/D column held by this lane

    v8f acc_o = {};
    v8f acc_m = {};

    const float* hrow = hist + (size_t)arow * BINS;

    for (int kb = 0; kb < BINS; kb += 4) {
        const int k0 = kb + hi * 2;
        const float h0 = hrow[k0];
        const float h1 = hrow[k0 + 1];
        v2f a_h = { h0, h1 };
        v2f a_m = { h0 * (float)k0, h1 * (float)(k0 + 1) };
        // U[k][n] = (k <= n) ? 1 : 0, generated in-register
        v2f b   = { (k0     <= bcol) ? 1.0f : 0.0f,
                    (k0 + 1 <= bcol) ? 1.0f : 0.0f };

        acc_o = __builtin_amdgcn_wmma_f32_16x16x4_f32(
            false, a_h, false, b, (short)0, acc_o, false, false);
        acc_m = __builtin_amdgcn_wmma_f32_16x16x4_f32(
            false, a_m, false, b, (short)0, acc_m, false, false);
    }

    const float invN = 1.0f / (float)NELEM;
    #pragma unroll
    for (int v = 0; v < 8; ++v) {
        const int row = m0 + v + hi * 8;       // C/D layout: lanes 16-31 hold M=v+8
        omega[(size_t)row * BINS + bcol] = acc_o[v] * invN;
        mu[(size_t)row * BINS + bcol]    = acc_m[v] * invN;
    }
}

// ---------------------------------------------------------------------------
// Kernel 3: per-row sigma_b^2 + first-occurrence argmax -> threshold.
// ---------------------------------------------------------------------------
__global__ void otsu_thr(const float* __restrict__ omega,
                         const float* __restrict__ mu,
                         const float* __restrict__ mnmx,
                         float* __restrict__ thr)   // [512]
{
    const int rid = blockIdx.x;      // 0..511
    const int j   = threadIdx.x;     // 0..255

    const float om   = omega[(size_t)rid * BINS + j];
    const float m    = mu[(size_t)rid * BINS + j];
    const float mu_t = mu[(size_t)rid * BINS + (BINS - 1)];

    const float num = mu_t * om - m;
    const float sb  = (num * num) / (om * (1.0f - om) + 1e-8f);

    __shared__ float sv[256];
    __shared__ int   si[256];
    sv[j] = sb; si[j] = j;
    __syncthreads();
    for (int s = 128; s > 0; s >>= 1) {
        if (j < s) {
            const float v2 = sv[j + s];
            const int   i2 = si[j + s];
            if (v2 > sv[j] || (v2 == sv[j] && i2 < si[j])) { sv[j] = v2; si[j] = i2; }
        }
        __syncthreads();
    }
    if (j == 0) {
        const float mn = mnmx[rid * 2 + 0];
        const float mx = mnmx[rid * 2 + 1];
        const float k  = (float)si[0];
        thr[rid] = mn + (mx - mn) * (k / 255.0f);
    }
}

// ---------------------------------------------------------------------------
// Kernel 4: elementwise masks, float4 streaming.
// out[0 .. B*N)      = new_mask       = (a > t_now) & (b <= t_past)
// out[B*N .. 2*B*N)  = disappear_mask = (b > t_past) & (a <= t_now)
// ---------------------------------------------------------------------------
__global__ void mask_kernel(const float* __restrict__ A,
                            const float* __restrict__ Bm,
                            const float* __restrict__ thr,
                            float* __restrict__ out)
{
    const int row = blockIdx.y;                                  // 0..255
    const int i   = blockIdx.x * blockDim.x + threadIdx.x;       // 0..NV4-1
    const float t_now  = thr[row];
    const float t_past = thr[256 + row];

    const float4* a4 = (const float4*)(A  + (size_t)row * NELEM);
    const float4* b4 = (const float4*)(Bm + (size_t)row * NELEM);
    const float4 a = a4[i];
    const float4 b = b4[i];

    float4 nm, dm;
    nm.x = (a.x > t_now  && b.x <= t_past) ? 1.0f : 0.0f;
    nm.y = (a.y > t_now  && b.y <= t_past) ? 1.0f : 0.0f;
    nm.z = (a.z > t_now  && b.z <= t_past) ? 1.0f : 0.0f;
    nm.w = (a.w > t_now  && b.w <= t_past) ? 1.0f : 0.0f;
    dm.x = (b.x > t_past && a.x <= t_now ) ? 1.0f : 0.0f;
    dm.y = (b.y > t_past && a.y <= t_now ) ? 1.0f : 0.0f;
    dm.z = (b.z > t_past && a.z <= t_now ) ? 1.0f : 0.0f;
    dm.w = (b.w > t_past && a.w <= t_now ) ? 1.0f : 0.0f;

    float4* outn = (float4*)(out + (size_t)row * NELEM);
    float4* outd = (float4*)(out + (size_t)BNUM * NELEM + (size_t)row * NELEM);
    outn[i] = nm;
    outd[i] = dm;
}

// ---------------------------------------------------------------------------
extern "C" void kernel_launch(void* const* d_in, const int* in_sizes, int n_in,
                              void* d_out, int out_size, void* d_ws, size_t ws_size,
                              hipStream_t stream) {
    const float* A  = (const float*)d_in[0];   // C_now2past [256,65536]
    const float* Bm = (const float*)d_in[1];   // C_past2now [256,65536]
    float* out = (float*)d_out;

    // workspace layout (floats)
    float* ws    = (float*)d_ws;
    float* mnmx  = ws;                              // 512*2      = 1024
    float* hist  = mnmx + NROWS_TOT * 2;            // 512*256    = 131072
    float* omega = hist + NROWS_TOT * BINS;         // 512*256
    float* mu    = omega + NROWS_TOT * BINS;        // 512*256
    float* thr   = mu + NROWS_TOT * BINS;           // 512

    // 1) min/max + histogram per row
    row_minmax_hist<<<dim3(NROWS_TOT), dim3(256), 0, stream>>>(A, Bm, mnmx, hist);

    // 2) WMMA cumulative sums (omega, mu)
    otsu_cumsum_wmma<<<dim3(BINS / 16, NROWS_TOT / 16), dim3(32), 0, stream>>>(hist, omega, mu);

    // 3) per-row Otsu threshold
    otsu_thr<<<dim3(NROWS_TOT), dim3(256), 0, stream>>>(omega, mu, mnmx, thr);

    // 4) elementwise masks
    mask_kernel<<<dim3(NV4 / 256, BNUM), dim3(256), 0, stream>>>(A, Bm, thr, out);
}